// AudioAttentionPooler_33105607918115
// MI455X (gfx1250) — compile-verified
//
#include <hip/hip_runtime.h>
#include <math.h>

#define HIDDEN 1024
#define NUM_HEADS 16
#define HEAD_DIM 64
#define PROJ 1024
#define B_ 32
#define T_ 2048

typedef __attribute__((ext_vector_type(2))) float v2f;
typedef __attribute__((ext_vector_type(8))) float v8f;

// ---------------------------------------------------------------------------
// Kernel A: wqT[i][h] = scale * sum_d Wk[i, h*64+d] * q[h*64+d]   (1024x16)
//           cq[h]    = scale * sum_d bk[h*64+d]   * q[h*64+d]
// ---------------------------------------------------------------------------
__global__ void build_wq_kernel(const float* __restrict__ kv_w,
                                const float* __restrict__ kv_b,
                                const float* __restrict__ query,
                                float* __restrict__ wqT,
                                float* __restrict__ cq) {
  const float scale = 0.125f;  // 1/sqrt(64)
  int i = blockIdx.x * blockDim.x + threadIdx.x;  // 0..1023
  const float* wrow = kv_w + (size_t)i * (2 * HIDDEN);
  for (int h = 0; h < NUM_HEADS; ++h) {
    float s = 0.f;
    const float* qh = query + h * HEAD_DIM;
    const float* wh = wrow + h * HEAD_DIM;
#pragma unroll 8
    for (int d = 0; d < HEAD_DIM; ++d) s += wh[d] * qh[d];
    wqT[i * NUM_HEADS + h] = s * scale;
  }
  if (blockIdx.x == 0 && threadIdx.x < NUM_HEADS) {
    int h = threadIdx.x;
    float s = 0.f;
    for (int d = 0; d < HEAD_DIM; ++d)
      s += kv_b[h * HEAD_DIM + d] * query[h * HEAD_DIM + d];
    cq[h] = s * scale;
  }
}

// ---------------------------------------------------------------------------
// Kernel B: logits[b][h][t] = hidden[b,t,:] . wq[h,:] + cq[h]
// One wave per 16-row t-tile. M=t (16), N=head (16), K=HIDDEN via f32 WMMA.
// wqT staged in 64KB LDS, reused by all 8 waves of the block.
// ---------------------------------------------------------------------------
__global__ void __launch_bounds__(256)
logits_wmma_kernel(const float* __restrict__ hidden,
                   const float* __restrict__ wqT,
                   const float* __restrict__ cq,
                   float* __restrict__ logits) {
  __shared__ float lds_wq[HIDDEN * NUM_HEADS];  // 64 KB
  for (int i = threadIdx.x; i < HIDDEN * NUM_HEADS; i += 256) lds_wq[i] = wqT[i];
  __syncthreads();

  const int wave = threadIdx.x >> 5;
  const int lane = threadIdx.x & 31;
  const int lm = lane & 15;   // A: row M ; B: col N ; D: col N
  const int lh = lane >> 4;   // half-wave select (K pair / M+8)
  const int tile = blockIdx.x * 8 + wave;     // 0..4095
  const int b = tile / (T_ / 16);
  const int t0 = (tile % (T_ / 16)) * 16;

  // A-tile: hidden rows t0..t0+15, K pair starting at 2*lh
  const float* arow = hidden + ((size_t)b * T_ + t0 + lm) * HIDDEN + 2 * lh;

  v8f acc = {};
  for (int k0 = 0; k0 < HIDDEN; k0 += 4) {
    v2f A, Bv;
    float2 a2 = *(const float2*)(arow + k0);   // K = k0+2*lh, +1 (contiguous)
    A[0] = a2.x; A[1] = a2.y;
    const int kb = k0 + 2 * lh;
    Bv[0] = lds_wq[kb * NUM_HEADS + lm];       // B[k][n=head]
    Bv[1] = lds_wq[(kb + 1) * NUM_HEADS + lm];
    acc = __builtin_amdgcn_wmma_f32_16x16x4_f32(
        false, A, false, Bv, (short)0, acc, false, false);
  }

  // D: M(t) = r + 8*lh, N(head) = lm
  const float c = cq[lm];
#pragma unroll
  for (int r = 0; r < 8; ++r) {
    int t = t0 + r + 8 * lh;
    logits[((size_t)b * NUM_HEADS + lm) * T_ + t] = acc[r] + c;
  }
}

// ---------------------------------------------------------------------------
// Kernel C: masked softmax over T for each (b,h) row, in place.
// ---------------------------------------------------------------------------
__global__ void __launch_bounds__(256)
softmax_kernel(float* __restrict__ attn, const int* __restrict__ mask) {
  __shared__ float red[256];
  const int row = blockIdx.x;  // b*16 + h
  const int b = row >> 4;
  float* rp = attn + (size_t)row * T_;
  const int* mp = mask + (size_t)b * T_;
  const int tid = threadIdx.x;

  float xs[T_ / 256];
  float m = -INFINITY;
  int i = 0;
  for (int t = tid; t < T_; t += 256, ++i) {
    float x = (mp[t] != 0) ? rp[t] : -INFINITY;
    xs[i] = x;
    m = fmaxf(m, x);
  }
  red[tid] = m;
  __syncthreads();
  for (int s = 128; s > 0; s >>= 1) {
    if (tid < s) red[tid] = fmaxf(red[tid], red[tid + s]);
    __syncthreads();
  }
  m = red[0];
  __syncthreads();

  float ev[T_ / 256];
  float sum = 0.f;
  i = 0;
  for (int t = tid; t < T_; t += 256, ++i) {
    float e = (xs[i] == -INFINITY) ? 0.f : __expf(xs[i] - m);
    ev[i] = e;
    sum += e;
  }
  red[tid] = sum;
  __syncthreads();
  for (int s = 128; s > 0; s >>= 1) {
    if (tid < s) red[tid] += red[tid + s];
    __syncthreads();
  }
  const float inv = 1.f / red[0];
  i = 0;
  for (int t = tid; t < T_; t += 256, ++i) rp[t] = ev[i] * inv;
}

// ---------------------------------------------------------------------------
// Kernel D: pooled[b] (16x1024) = attn[b] (16xT) @ hidden[b] (Tx1024)
// One wave per (b, 16-wide n-tile). M=head, K=t, N=hidden-dim via f32 WMMA.
// This is the second (and last) full pass over hidden_states.
// ---------------------------------------------------------------------------
__global__ void __launch_bounds__(256)
pool_wmma_kernel(const float* __restrict__ attn,
                 const float* __restrict__ hidden,
                 float* __restrict__ pooled) {
  const int wave = threadIdx.x >> 5;
  const int lane = threadIdx.x & 31;
  const int lm = lane & 15;
  const int lh = lane >> 4;
  const int w = blockIdx.x * 8 + wave;  // 0..2047
  const int b = w >> 6;
  const int n0 = (w & 63) * 16;

  const float* ap = attn + ((size_t)b * NUM_HEADS + lm) * T_ + 2 * lh;  // A row = head lm
  const float* hp = hidden + (size_t)b * T_ * HIDDEN + n0 + lm;         // B col = n0+lm

  v8f acc = {};
  for (int t0 = 0; t0 < T_; t0 += 4) {
    v2f A, Bv;
    float2 a2 = *(const float2*)(ap + t0);     // attn[b][lm][t0+2*lh .. +1], contiguous
    A[0] = a2.x; A[1] = a2.y;
    const int tb = t0 + 2 * lh;
    Bv[0] = hp[(size_t)tb * HIDDEN];           // hidden[b][tb  ][n0+lm]
    Bv[1] = hp[(size_t)(tb + 1) * HIDDEN];     // hidden[b][tb+1][n0+lm]
    acc = __builtin_amdgcn_wmma_f32_16x16x4_f32(
        false, A, false, Bv, (short)0, acc, false, false);
  }

  // D: M(head) = r + 8*lh, N = n0 + lm
#pragma unroll
  for (int r = 0; r < 8; ++r) {
    int h = r + 8 * lh;
    pooled[((size_t)b * NUM_HEADS + h) * HIDDEN + n0 + lm] = acc[r];
  }
}

// ---------------------------------------------------------------------------
// Kernel E: z[b][j] = pooled[b][j/64][:] . Wv[:,j] + bv[j]
//   Wv[i][j] = kv_w[i*2048 + 1024 + j], bv[j] = kv_b[1024+j]
// ---------------------------------------------------------------------------
__global__ void __launch_bounds__(256)
attn_out_kernel(const float* __restrict__ pooled,
                const float* __restrict__ kv_w,
                const float* __restrict__ kv_b,
                float* __restrict__ z) {
  const int b = blockIdx.x;
  const int j = blockIdx.y * 256 + threadIdx.x;  // 0..1023
  const float* pr = pooled + ((size_t)b * NUM_HEADS + (j >> 6)) * HIDDEN;
  float acc = kv_b[HIDDEN + j];
#pragma unroll 4
  for (int i = 0; i < HIDDEN; ++i)
    acc += pr[i] * kv_w[(size_t)i * (2 * HIDDEN) + HIDDEN + j];
  z[(size_t)b * HIDDEN + j] = acc;
}

// ---------------------------------------------------------------------------
// Kernel F: out[b][p] = z[b][:] . out_w[:,p] + out_b[p]
// ---------------------------------------------------------------------------
__global__ void __launch_bounds__(256)
final_proj_kernel(const float* __restrict__ z,
                  const float* __restrict__ out_w,
                  const float* __restrict__ out_b,
                  float* __restrict__ out) {
  const int b = blockIdx.x;
  const int p = blockIdx.y * 256 + threadIdx.x;  // 0..1023
  const float* zr = z + (size_t)b * HIDDEN;
  float acc = out_b[p];
#pragma unroll 4
  for (int j = 0; j < HIDDEN; ++j)
    acc += zr[j] * out_w[(size_t)j * PROJ + p];
  out[(size_t)b * PROJ + p] = acc;
}

// ---------------------------------------------------------------------------
extern "C" void kernel_launch(void* const* d_in, const int* in_sizes, int n_in,
                              void* d_out, int out_size, void* d_ws, size_t ws_size,
                              hipStream_t stream) {
  const float* hidden = (const float*)d_in[0];
  const int*   mask   = (const int*)d_in[1];
  const float* kv_w   = (const float*)d_in[2];
  const float* kv_b   = (const float*)d_in[3];
  const float* out_w  = (const float*)d_in[4];
  const float* out_b  = (const float*)d_in[5];
  const float* query  = (const float*)d_in[6];
  float* out = (float*)d_out;

  char* ws = (char*)d_ws;
  float* wqT    = (float*)(ws);                                   // 64 KB
  float* cq     = (float*)(ws + 64 * 1024);                       // 64 B
  float* attn   = (float*)(ws + 128 * 1024);                      // 4 MB (logits -> probs, in place)
  float* pooled = (float*)(ws + 128 * 1024 + 4 * 1024 * 1024);    // 2 MB
  float* z      = (float*)(ws + 128 * 1024 + 6 * 1024 * 1024);    // 128 KB

  build_wq_kernel<<<4, 256, 0, stream>>>(kv_w, kv_b, query, wqT, cq);
  logits_wmma_kernel<<<(B_ * T_ / 16) / 8, 256, 0, stream>>>(hidden, wqT, cq, attn);
  softmax_kernel<<<B_ * NUM_HEADS, 256, 0, stream>>>(attn, mask);
  pool_wmma_kernel<<<(B_ * 64) / 8, 256, 0, stream>>>(attn, hidden, pooled);
  attn_out_kernel<<<dim3(B_, HIDDEN / 256), 256, 0, stream>>>(pooled, kv_w, kv_b, z);
  final_proj_kernel<<<dim3(B_, PROJ / 256), 256, 0, stream>>>(z, out_w, out_b, out);
}